// UMAPLoss_3118146257316
// MI455X (gfx1250) — compile-verified
//
#include <hip/hip_runtime.h>

// UMAP loss, fused for MI455X (gfx1250, wave32, WMMA, async-to-LDS).
//
// Roofline: Gram matrix is ~17 GFLOP (trivial for f32 WMMA 16x16x4); the
// 67M-pair transcendental pass dominates; X (4MB) is L2-resident so tiles
// are re-read instead of materializing the 256MB distance matrix.
// top_k adjacency reduced to a per-row 14th-smallest threshold, so one
// block owns 16 rows end-to-end: phase 1 = threshold, phase 2 = loss.
// B super-tiles are double-buffered via GLOBAL_LOAD_ASYNC_TO_LDS_B128
// (ASYNCcnt); tile loop is unrolled x2 so buffers bind statically, and the
// k-loop is software-pipelined so ds_loads overlap v_wmma issue.

#define N       8192
#define D       128
#define K_NEI   14
#define ROWS    16      // rows per block (one A tile)
#define COLS    128     // columns per super-tile (8 waves x 16)
#define NTILES  (N / COLS)
#define LDA     132     // padded LDS row stride (bank-conflict free)
#define THREADS 256

typedef __attribute__((ext_vector_type(2))) float v2f;
typedef __attribute__((ext_vector_type(8))) float v8f;
typedef __attribute__((ext_vector_type(4))) int   v4i;

typedef v4i __attribute__((address_space(1)))* gv4i_p;  // global b128
typedef v4i __attribute__((address_space(3)))* lv4i_p;  // LDS b128

#if __has_builtin(__builtin_amdgcn_global_load_async_to_lds_b128)
#define HAS_ASYNC 1
#else
#define HAS_ASYNC 0
#endif

// 16-byte global -> LDS copy: async DMA path on CDNA5, sync fallback else.
__device__ __forceinline__ void cp16(const float* g, float* l) {
#if HAS_ASYNC
  __builtin_amdgcn_global_load_async_to_lds_b128((gv4i_p)g, (lv4i_p)l, 0, 0);
#else
  const float4 v = *(const float4*)g;
  l[0] = v.x; l[1] = v.y; l[2] = v.z; l[3] = v.w;
#endif
}

__device__ __forceinline__ void async_wait0() {
#if HAS_ASYNC
#if __has_builtin(__builtin_amdgcn_s_wait_asynccnt)
  __builtin_amdgcn_s_wait_asynccnt(0);
#else
  asm volatile("s_wait_asynccnt 0" ::: "memory");
#endif
#endif
}

// 16x16 Gram tile via f32 WMMA, K=128, one-stage pipelined fragment loads.
__device__ __forceinline__ v8f gram_tile(const float* ap, const float* bp) {
  v8f c = {0.f, 0.f, 0.f, 0.f, 0.f, 0.f, 0.f, 0.f};
  v2f a, b;
  a[0] = ap[0]; a[1] = ap[1];
  b[0] = bp[0]; b[1] = bp[1];
#pragma unroll
  for (int k0 = 4; k0 < D; k0 += 4) {
    v2f an, bn;                          // prefetch next fragments first
    an[0] = ap[k0]; an[1] = ap[k0 + 1];
    bn[0] = bp[k0]; bn[1] = bp[k0 + 1];
    c = __builtin_amdgcn_wmma_f32_16x16x4_f32(false, a, false, b,
                                              (short)0, c, false, false);
    a = an; b = bn;
  }
  c = __builtin_amdgcn_wmma_f32_16x16x4_f32(false, a, false, b,
                                            (short)0, c, false, false);
  return c;
}

// ---------------- row squared norms: one wave32 per row ----------------
__global__ __launch_bounds__(THREADS) void sqnorm_kernel(
    const float* __restrict__ X, float* __restrict__ sq) {
  const int lane = threadIdx.x & 31;
  const int wv   = threadIdx.x >> 5;
  const int row  = blockIdx.x * (THREADS / 32) + wv;
  const float* xp = X + (size_t)row * D;
  float s = 0.f;
#pragma unroll
  for (int i = 0; i < D / 32; ++i) { const float v = xp[lane + 32 * i]; s += v * v; }
#pragma unroll
  for (int off = 16; off > 0; off >>= 1) s += __shfl_xor(s, off, 32);
  if (lane == 0) sq[row] = s;
}

// ---------------- fused threshold + loss kernel ----------------
__global__ __launch_bounds__(THREADS) void umap_main(
    const float* __restrict__ X, const float* __restrict__ LO,
    const float* __restrict__ sq, float* __restrict__ partial) {
  extern __shared__ float smem[];
  float* As  = smem;                 // [16][132] A row-stripe
  float* Bs0 = As + ROWS * LDA;      // [128][132] B super-tile, buffer 0
  float* Bs1 = Bs0 + COLS * LDA;     // [128][132] B super-tile, buffer 1
  float* Ds  = Bs1 + COLS * LDA;     // [16][132] distance chunk (phase 1)
  float* SQR = Ds + ROWS * LDA;      // [16] row sq-norms
  float* THR = SQR + ROWS;           // [16] per-row 14th-smallest distance
  float* LOR = THR + ROWS;           // [16][2] row low-dim coords
  float* RED = LOR + 2 * ROWS;       // [8] per-wave partial sums

  const int tid  = threadIdx.x;
  const int lane = tid & 31;
  const int wv   = tid >> 5;
  const int half = lane >> 4;        // 0: lanes 0-15, 1: lanes 16-31
  const int l15  = lane & 15;
  const int kb   = half * 2;         // K sub-offset per ISA f32 A/B layout
  const int r0   = blockIdx.x * ROWS;
  const int boff = (wv * 16 + l15) * LDA + kb;  // B-fragment offset in a buffer

  // issue async fill of one B super-tile (16 x b128 per thread, coalesced)
  auto issue_tile = [&](int c0, float* buf) {
    for (int f = tid; f < COLS * (D / 4); f += THREADS) {
      const int r = f >> 5, k4 = f & 31;
      cp16(X + (size_t)(c0 + r) * D + k4 * 4, buf + r * LDA + k4 * 4);
    }
  };

  // load A tile (16 x 128) once, padded
  for (int f = tid; f < ROWS * (D / 4); f += THREADS) {
    const int r = f >> 5, k4 = f & 31;
    const float4 v = ((const float4*)(X + (size_t)(r0 + r) * D))[k4];
    float* dst = As + r * LDA + k4 * 4;
    dst[0] = v.x; dst[1] = v.y; dst[2] = v.z; dst[3] = v.w;
  }
  if (tid < ROWS) {
    SQR[tid]         = sq[r0 + tid];
    LOR[2 * tid]     = LO[2 * (r0 + tid)];
    LOR[2 * tid + 1] = LO[2 * (r0 + tid) + 1];
  }
  issue_tile(0, Bs0);   // prefetch tile 0 while A settles

  // fragment base pointer for A (ISA 7.12.2: f32 16x4 A layout)
  const float* ap = As + l15 * LDA + kb;

  // per-row top-14 smallest (register list), threads 0..15 own one row each
  float topv[K_NEI];
#pragma unroll
  for (int q = 0; q < K_NEI; ++q) topv[q] = 3.0e38f;
  float worst = 3.0e38f;

  // ================= phase 1: kth-smallest threshold =================
  auto p1_step = [&](int t, const float* cbuf, float* pbuf) {
    const int c0 = t * COLS;
    async_wait0();
    __syncthreads();                     // tile t resident in all waves
    if (t + 1 < NTILES) issue_tile((t + 1) * COLS, pbuf);

    const v8f c = gram_tile(ap, cbuf + boff);
    const float sc = sq[c0 + wv * 16 + l15];
#pragma unroll
    for (int v = 0; v < 8; ++v) {
      const int m = v + 8 * half;        // C-tile row (ISA layout)
      const float d2 = SQR[m] + sc - 2.0f * c[v];
      Ds[m * LDA + wv * 16 + l15] = sqrtf(fmaxf(d2, 0.f));
    }
    __syncthreads();

    if (tid < ROWS) {                    // row-owner scan of Ds
      for (int j = 0; j < COLS; ++j) {
        const float d = Ds[tid * LDA + j];
        if (d < worst) {                 // ~k*ln(N) hits total
          int wi = 0; float wvv = topv[0];
#pragma unroll
          for (int q = 1; q < K_NEI; ++q) if (topv[q] > wvv) { wvv = topv[q]; wi = q; }
#pragma unroll
          for (int q = 0; q < K_NEI; ++q) if (q == wi) topv[q] = d;
          wvv = topv[0];
#pragma unroll
          for (int q = 1; q < K_NEI; ++q) wvv = fmaxf(wvv, topv[q]);
          worst = wvv;
        }
      }
    }
    // scan reads Ds only; the concurrent async fill targets the other B
    // buffer, and the next Ds rewrite is gated by the top barrier.
  };
  for (int t = 0; t < NTILES; t += 2) {  // static buffer binding per parity
    p1_step(t,     Bs0, Bs1);
    p1_step(t + 1, Bs1, Bs0);
  }
  if (tid < ROWS) THR[tid] = worst;
  __syncthreads();
  issue_tile(0, Bs0);                    // prefetch for phase 2

  // ================= phase 2: fused loss =================
  float fsum = 0.f;
  auto p2_step = [&](int t, const float* cbuf, float* pbuf) {
    const int c0 = t * COLS;
    async_wait0();
    __syncthreads();                     // tile t resident; other buffer free
    if (t + 1 < NTILES) issue_tile((t + 1) * COLS, pbuf);

    const v8f c = gram_tile(ap, cbuf + boff);
    const int   coln = c0 + wv * 16 + l15;
    const float sc   = sq[coln];
    const float lcx  = LO[2 * coln], lcy = LO[2 * coln + 1];
#pragma unroll
    for (int v = 0; v < 8; ++v) {
      const int   m  = v + 8 * half;
      const float dh = sqrtf(fmaxf(SQR[m] + sc - 2.0f * c[v], 0.f));
      const float dx = LOR[2 * m] - lcx, dy = LOR[2 * m + 1] - lcy;
      const float dl = sqrtf(dx * dx + dy * dy);
      const float hi = __expf(-dh);      // SPREAD = 1
      const float ls = 1.f / (1.f + dl);
      const float adj = (dh <= THR[m]) ? 1.f : 0.f;
      fsum += adj * hi * __logf(ls + 1e-10f)
            + (1.f - adj) * (1.f - hi) * __logf(1.f - ls + 1e-10f);
    }
  };
  for (int t = 0; t < NTILES; t += 2) {
    p2_step(t,     Bs0, Bs1);
    p2_step(t + 1, Bs1, Bs0);
  }

  // block reduction (wave32 shuffle + LDS), deterministic partials
#pragma unroll
  for (int off = 16; off > 0; off >>= 1) fsum += __shfl_xor(fsum, off, 32);
  if (lane == 0) RED[wv] = fsum;
  __syncthreads();
  if (tid == 0) {
    float s = 0.f;
#pragma unroll
    for (int i = 0; i < THREADS / 32; ++i) s += RED[i];
    partial[blockIdx.x] = s;
  }
}

// ---------------- final reduction: 512 partials -> scalar ----------------
__global__ __launch_bounds__(256) void umap_reduce(
    const float* __restrict__ partial, float* __restrict__ out) {
  __shared__ float red[256];
  float s = 0.f;
  for (int i = threadIdx.x; i < N / ROWS; i += 256) s += partial[i];
  red[threadIdx.x] = s;
  __syncthreads();
  for (int off = 128; off > 0; off >>= 1) {
    if (threadIdx.x < off) red[threadIdx.x] += red[threadIdx.x + off];
    __syncthreads();
  }
  if (threadIdx.x == 0) out[0] = -100.0f * (red[0] / ((float)N * (float)N));
}

extern "C" void kernel_launch(void* const* d_in, const int* in_sizes, int n_in,
                              void* d_out, int out_size, void* d_ws, size_t ws_size,
                              hipStream_t stream) {
  const float* X  = (const float*)d_in[0];   // [8192,128] f32
  const float* LO = (const float*)d_in[1];   // [8192,2]   f32
  float* out      = (float*)d_out;           // scalar f32
  float* sq       = (float*)d_ws;            // [8192] row sq-norms
  float* partial  = sq + N;                  // [512] block partial sums

  sqnorm_kernel<<<N / (THREADS / 32), THREADS, 0, stream>>>(X, sq);

  const size_t shmem =
      (size_t)(ROWS * LDA + 2 * COLS * LDA + ROWS * LDA + ROWS + ROWS +
               2 * ROWS + THREADS / 32) * sizeof(float);  // ~152 KB < 320 KB/WGP
  umap_main<<<N / ROWS, THREADS, shmem, stream>>>(X, LO, sq, partial);

  umap_reduce<<<1, 256, 0, stream>>>(partial, out);
}